// ExtendedLightGCNModel_89343909691632
// MI455X (gfx1250) — compile-verified
//
#include <hip/hip_runtime.h>

#define NUM_USERS  100000
#define NUM_ITEMS  50000
#define NNZ        2400000
#define EMB        64
#define NTOT       (NUM_USERS + NUM_ITEMS)
#define NPAIR      16384
#define NUM_LAYERS 3

typedef __attribute__((ext_vector_type(2))) float v2f;
typedef __attribute__((ext_vector_type(8))) float v8f;

// ---------------------------------------------------------------------------
// init: cur = concat(emb_user, emb_item); acc = cur; nxt = 0   (float4 wide)
// ---------------------------------------------------------------------------
__global__ void lgcn_init(const float* __restrict__ eu, const float* __restrict__ ei,
                          float* __restrict__ cur, float* __restrict__ acc,
                          float* __restrict__ nxt) {
    size_t i = (size_t)blockIdx.x * blockDim.x + threadIdx.x;   // float4 index
    const size_t total4 = (size_t)NTOT * EMB / 4;
    if (i >= total4) return;
    const size_t user4 = (size_t)NUM_USERS * EMB / 4;
    float4 v = (i < user4) ? ((const float4*)eu)[i]
                           : ((const float4*)ei)[i - user4];
    ((float4*)cur)[i] = v;
    ((float4*)acc)[i] = v;
    ((float4*)nxt)[i] = make_float4(0.f, 0.f, 0.f, 0.f);
}

// ---------------------------------------------------------------------------
// SpMM: nxt[row[e]] += val[e] * cur[col[e]]
// 16 lanes cooperate per edge, each moving a float4 (16*16B = full 64-f32 row).
// Gathers are 128-bit coalesced; scatters are global_atomic_add_f32 (L2).
// ---------------------------------------------------------------------------
__global__ void lgcn_spmm(const int* __restrict__ row, const int* __restrict__ col,
                          const float* __restrict__ val,
                          const float* __restrict__ cur, float* __restrict__ nxt) {
    long long t = (long long)blockIdx.x * blockDim.x + threadIdx.x;
    long long e = t >> 4;
    int lane = (int)(t & 15);
    if (e >= NNZ) return;
    int   r = row[e];
    int   c = col[e];
    float w = val[e];
    float4 x = ((const float4*)(cur + (size_t)c * EMB))[lane];
    float* dst = nxt + (size_t)r * EMB + lane * 4;
    atomicAdd(dst + 0, w * x.x);
    atomicAdd(dst + 1, w * x.y);
    atomicAdd(dst + 2, w * x.z);
    atomicAdd(dst + 3, w * x.w);
}

// ---------------------------------------------------------------------------
// acc += layer   (float4 wide)
// ---------------------------------------------------------------------------
__global__ void lgcn_accum(float* __restrict__ acc, const float* __restrict__ layer) {
    size_t i = (size_t)blockIdx.x * blockDim.x + threadIdx.x;
    const size_t total4 = (size_t)NTOT * EMB / 4;
    if (i >= total4) return;
    float4 a = ((const float4*)acc)[i];
    float4 b = ((const float4*)layer)[i];
    a.x += b.x; a.y += b.y; a.z += b.z; a.w += b.w;
    ((float4*)acc)[i] = a;
}

// ---------------------------------------------------------------------------
// zero a buffer (float4 wide)
// ---------------------------------------------------------------------------
__global__ void lgcn_zero(float* __restrict__ p) {
    size_t i = (size_t)blockIdx.x * blockDim.x + threadIdx.x;
    const size_t total4 = (size_t)NTOT * EMB / 4;
    if (i >= total4) return;
    ((float4*)p)[i] = make_float4(0.f, 0.f, 0.f, 0.f);
}

// ---------------------------------------------------------------------------
// Final scoring via V_WMMA_F32_16X16X4_F32.
// One wave32 per tile of 16 pairs: D = U(16x64) x V^T(64x16) over 16 K-steps,
// gamma = diag(D) * 1/16  (folds the /(L+1)=/4 mean applied to both factors).
//
// Fragment layout (ISA 7.12.2, 32-bit A 16x4): lane l holds row m=l&15,
// K = (l>>4)*2 and +1.  B 4x16: lane l holds col n=l&15, same K split.
// D (16x16 f32, 8 VGPRs): lanes 0-15 -> N=lane, M=vgpr; lanes 16-31 -> M=vgpr+8.
// ---------------------------------------------------------------------------
__global__ __launch_bounds__(32) void lgcn_gamma(const float* __restrict__ acc,
                                                 const int* __restrict__ users,
                                                 const int* __restrict__ items,
                                                 float* __restrict__ out) {
    int tile = blockIdx.x;
    int l    = threadIdx.x;          // 0..31, full wave, EXEC all ones
    int m    = l & 15;
    int kb   = (l >> 4) * 2;

    int u  = users[tile * 16 + m];
    int it = items[tile * 16 + m];
    const float* urow = acc + (size_t)u * EMB;
    const float* vrow = acc + ((size_t)NUM_USERS + (size_t)it) * EMB;

    v8f d = {};
#pragma unroll
    for (int k0 = 0; k0 < EMB; k0 += 4) {
        v2f a = *(const v2f*)(urow + k0 + kb);   // 8B-aligned: k0+kb is even
        v2f b = *(const v2f*)(vrow + k0 + kb);
        d = __builtin_amdgcn_wmma_f32_16x16x4_f32(
                /*neg_a=*/false, a, /*neg_b=*/false, b,
                /*c_mod=*/(short)0, d, /*reuse_a=*/false, /*reuse_b=*/false);
    }

    // Extract diagonal: D[j][j] at (lane j, vgpr j) for j<8,
    // D[8+j][8+j] at (lane 24+j, vgpr j) for j<8.
    const float scale = 1.0f / 16.0f;            // (1/4)*(1/4) from layer mean
    float g = 0.f; int idx = -1;
#pragma unroll
    for (int j = 0; j < 8; ++j) {
        if (l == j)      { g = d[j]; idx = j;     }
        if (l == 24 + j) { g = d[j]; idx = 8 + j; }
    }
    if (idx >= 0) out[tile * 16 + idx] = g * scale;
}

// ---------------------------------------------------------------------------
extern "C" void kernel_launch(void* const* d_in, const int* in_sizes, int n_in,
                              void* d_out, int out_size, void* d_ws, size_t ws_size,
                              hipStream_t stream) {
    const int*   row      = (const int*)  d_in[0];
    const int*   col      = (const int*)  d_in[1];
    const float* val      = (const float*)d_in[2];
    const float* emb_user = (const float*)d_in[3];
    const float* emb_item = (const float*)d_in[4];
    const int*   users    = (const int*)  d_in[5];
    const int*   items    = (const int*)  d_in[6];
    float*       out      = (float*)d_out;
    (void)in_sizes; (void)n_in; (void)out_size; (void)ws_size;

    const size_t matElems = (size_t)NTOT * EMB;
    float* cur = (float*)d_ws;
    float* nxt = cur + matElems;
    float* acc = nxt + matElems;

    const size_t total4 = matElems / 4;
    const int    tpb    = 256;
    const int    gElem  = (int)((total4 + tpb - 1) / tpb);

    lgcn_init<<<gElem, tpb, 0, stream>>>(emb_user, emb_item, cur, acc, nxt);

    for (int layer = 0; layer < NUM_LAYERS; ++layer) {
        long long spmmThreads = (long long)NNZ * 16;
        int gSpmm = (int)((spmmThreads + tpb - 1) / tpb);
        lgcn_spmm<<<gSpmm, tpb, 0, stream>>>(row, col, val, cur, nxt);
        lgcn_accum<<<gElem, tpb, 0, stream>>>(acc, nxt);
        float* t = cur; cur = nxt; nxt = t;       // cur <- new layer
        if (layer + 1 < NUM_LAYERS)
            lgcn_zero<<<gElem, tpb, 0, stream>>>(nxt);
    }

    lgcn_gamma<<<NPAIR / 16, 32, 0, stream>>>(acc, users, items, out);
}